// Wavelet_76055280877840
// MI455X (gfx1250) — compile-verified
//
#include <hip/hip_runtime.h>
#include <stdint.h>

// ---------------------------------------------------------------------------
// 4-level fused Haar wavelet pyramid, 32x3x512x512 fp32.
// One 256-thread block (8 wave32) per 64x64 tile of one (batch,channel).
// Input tile staged to LDS via CDNA5 async global->LDS loads (ASYNCcnt),
// all cascade levels computed from LDS, outputs streamed with NT stores.
// ---------------------------------------------------------------------------

#define TILE       64
#define SIN_STRIDE 68   // 68*4 = 272 B, 16B aligned rows
#define S1_STRIDE  34
#define S2_STRIDE  18
#define S3_STRIDE  10

typedef int v4i __attribute__((ext_vector_type(4)));
typedef __attribute__((address_space(1))) v4i as1_v4i;  // global int4
typedef __attribute__((address_space(3))) v4i as3_v4i;  // LDS int4

#if __has_builtin(__builtin_amdgcn_global_load_async_to_lds_b128)
#define HAVE_ASYNC_LDS 1
#else
#define HAVE_ASYNC_LDS 0
#endif

__device__ __forceinline__ void stage16(const float* __restrict__ g, float* l) {
#if HAVE_ASYNC_LDS
  __builtin_amdgcn_global_load_async_to_lds_b128(
      (as1_v4i*)g, (as3_v4i*)l, /*offset=*/0, /*cpol=*/0);
#else
  *(float4*)l = *(const float4*)g;
#endif
}

__device__ __forceinline__ void stage_wait() {
#if HAVE_ASYNC_LDS
#if __has_builtin(__builtin_amdgcn_s_wait_asynccnt)
  __builtin_amdgcn_s_wait_asynccnt(0);
#else
  asm volatile("s_wait_asynccnt 0" ::: "memory");
#endif
#endif
}

// One 2x2 Haar butterfly matching the reference exactly:
// L=(r0+r1)/2, H=|r0-r1| over rows, then same over columns.
__device__ __forceinline__ void haar2x2(float a, float b, float c, float d,
                                        float& ll, float& lh, float& hl, float& hh) {
  float ly0 = 0.5f * (a + c);
  float ly1 = 0.5f * (b + d);
  float hy0 = fabsf(a - c);
  float hy1 = fabsf(b - d);
  ll = 0.5f * (ly0 + ly1);
  lh = fabsf(ly0 - ly1);
  hl = 0.5f * (hy0 + hy1);
  hh = fabsf(hy0 - hy1);
}

__global__ __launch_bounds__(256) void haar_pyramid4_kernel(
    const float* __restrict__ x,
    float* __restrict__ t1, float* __restrict__ t2,
    float* __restrict__ t3, float* __restrict__ t4) {
  __shared__ float s_in[TILE * SIN_STRIDE];  // 64x64 input tile (padded)
  __shared__ float s_l1[32 * S1_STRIDE];     // LL pyramid
  __shared__ float s_l2[16 * S2_STRIDE];
  __shared__ float s_l3[8 * S3_STRIDE];

  const int tid  = threadIdx.x;
  const int bid  = blockIdx.x;
  const int tile = bid & 63;          // 8x8 tiles per 512x512 image
  const int tx   = tile & 7;
  const int ty   = tile >> 3;
  const int c    = (bid >> 6) % 3;    // channel
  const int n    = bid / 192;         // batch

  const float* src =
      x + (((size_t)(n * 3 + c)) * 512 + (size_t)ty * TILE) * 512 + (size_t)tx * TILE;

  // ---- Phase A: stage 64x64 tile to LDS (async data mover, 16B per lane) ----
#pragma unroll
  for (int it = 0; it < 4; ++it) {
    int q     = tid + it * 256;   // 1024 chunks of 16B
    int row   = q >> 4;           // 16 chunks per 64-float row
    int chunk = q & 15;
    stage16(src + (size_t)row * 512 + chunk * 4, &s_in[row * SIN_STRIDE + chunk * 4]);
  }
  stage_wait();
  __syncthreads();

  const size_t chan = (size_t)(n * 12 + c * 4);  // [LL,LH,HL,HH] slot base

  // ---- Level 1: 32x32 outputs, 4 per thread ----
#pragma unroll
  for (int k = 0; k < 4; ++k) {
    int idx = tid + k * 256;
    int i = idx >> 5, j = idx & 31;
    const float2 r0 = *(const float2*)&s_in[(2 * i) * SIN_STRIDE + 2 * j];
    const float2 r1 = *(const float2*)&s_in[(2 * i + 1) * SIN_STRIDE + 2 * j];
    float ll, lh, hl, hh;
    haar2x2(r0.x, r0.y, r1.x, r1.y, ll, lh, hl, hh);
    s_l1[i * S1_STRIDE + j] = ll;
    size_t base = (chan * 256 + (size_t)(ty * 32 + i)) * 256 + (tx * 32 + j);
    __builtin_nontemporal_store(ll, &t1[base]);
    __builtin_nontemporal_store(lh, &t1[base + 1 * 256 * 256]);
    __builtin_nontemporal_store(hl, &t1[base + 2 * 256 * 256]);
    __builtin_nontemporal_store(hh, &t1[base + 3 * 256 * 256]);
  }
  __syncthreads();

  // ---- Level 2: 16x16 outputs, 1 per thread ----
  {
    int i = tid >> 4, j = tid & 15;
    const float2 r0 = *(const float2*)&s_l1[(2 * i) * S1_STRIDE + 2 * j];
    const float2 r1 = *(const float2*)&s_l1[(2 * i + 1) * S1_STRIDE + 2 * j];
    float ll, lh, hl, hh;
    haar2x2(r0.x, r0.y, r1.x, r1.y, ll, lh, hl, hh);
    s_l2[i * S2_STRIDE + j] = ll;
    size_t base = (chan * 128 + (size_t)(ty * 16 + i)) * 128 + (tx * 16 + j);
    __builtin_nontemporal_store(ll, &t2[base]);
    __builtin_nontemporal_store(lh, &t2[base + 1 * 128 * 128]);
    __builtin_nontemporal_store(hl, &t2[base + 2 * 128 * 128]);
    __builtin_nontemporal_store(hh, &t2[base + 3 * 128 * 128]);
  }
  __syncthreads();

  // ---- Level 3: 8x8 outputs, threads 0..63 ----
  if (tid < 64) {
    int i = tid >> 3, j = tid & 7;
    const float2 r0 = *(const float2*)&s_l2[(2 * i) * S2_STRIDE + 2 * j];
    const float2 r1 = *(const float2*)&s_l2[(2 * i + 1) * S2_STRIDE + 2 * j];
    float ll, lh, hl, hh;
    haar2x2(r0.x, r0.y, r1.x, r1.y, ll, lh, hl, hh);
    s_l3[i * S3_STRIDE + j] = ll;
    size_t base = (chan * 64 + (size_t)(ty * 8 + i)) * 64 + (tx * 8 + j);
    __builtin_nontemporal_store(ll, &t3[base]);
    __builtin_nontemporal_store(lh, &t3[base + 1 * 64 * 64]);
    __builtin_nontemporal_store(hl, &t3[base + 2 * 64 * 64]);
    __builtin_nontemporal_store(hh, &t3[base + 3 * 64 * 64]);
  }
  __syncthreads();

  // ---- Level 4: 4x4 outputs, threads 0..15.
  // Reference quirk: b4 = g4 -> green writes blue's t4 slots, blue skips t4.
  if (tid < 16 && c != 2) {
    int i = tid >> 2, j = tid & 3;
    const float2 r0 = *(const float2*)&s_l3[(2 * i) * S3_STRIDE + 2 * j];
    const float2 r1 = *(const float2*)&s_l3[(2 * i + 1) * S3_STRIDE + 2 * j];
    float ll, lh, hl, hh;
    haar2x2(r0.x, r0.y, r1.x, r1.y, ll, lh, hl, hh);
    size_t base = (chan * 32 + (size_t)(ty * 4 + i)) * 32 + (tx * 4 + j);
    __builtin_nontemporal_store(ll, &t4[base]);
    __builtin_nontemporal_store(lh, &t4[base + 1 * 32 * 32]);
    __builtin_nontemporal_store(hl, &t4[base + 2 * 32 * 32]);
    __builtin_nontemporal_store(hh, &t4[base + 3 * 32 * 32]);
    if (c == 1) {  // duplicate green into blue's 4 slots (chan +4 -> +4*32*32)
      size_t base2 = base + (size_t)4 * 32 * 32;
      __builtin_nontemporal_store(ll, &t4[base2]);
      __builtin_nontemporal_store(lh, &t4[base2 + 1 * 32 * 32]);
      __builtin_nontemporal_store(hl, &t4[base2 + 2 * 32 * 32]);
      __builtin_nontemporal_store(hh, &t4[base2 + 3 * 32 * 32]);
    }
  }
}

extern "C" void kernel_launch(void* const* d_in, const int* in_sizes, int n_in,
                              void* d_out, int out_size, void* d_ws, size_t ws_size,
                              hipStream_t stream) {
  (void)in_sizes; (void)n_in; (void)out_size; (void)d_ws; (void)ws_size;
  const float* x = (const float*)d_in[0];
  float* out = (float*)d_out;
  float* t1 = out;                                        // (32,12,256,256)
  float* t2 = t1 + (size_t)32 * 12 * 256 * 256;           // (32,12,128,128)
  float* t3 = t2 + (size_t)32 * 12 * 128 * 128;           // (32,12,64,64)
  float* t4 = t3 + (size_t)32 * 12 * 64 * 64;             // (32,12,32,32)

  dim3 grid(32 * 3 * 64);  // batch * channel * (8x8 tiles)
  dim3 block(256);
  hipLaunchKernelGGL(haar_pyramid4_kernel, grid, block, 0, stream, x, t1, t2, t3, t4);
}